// ECELoss_42958262894785
// MI455X (gfx1250) — compile-verified
//
#include <hip/hip_runtime.h>
#include <math.h>
#include <stdint.h>

#define NROWS 262144
#define NCOLS 1000
#define NBINS 15
#define WAVES_PER_BLOCK 8
#define BLOCK_THREADS (WAVES_PER_BLOCK * 32)
#define ROW_PAD 1024   // LDS slice per wave (floats), 4KB, b128-aligned

typedef float v4f __attribute__((ext_vector_type(4)));

// ---------------------------------------------------------------------------
// Kernel 1: zero the 45-float accumulator workspace (counts, sum_conf, sum_acc)
// ---------------------------------------------------------------------------
__global__ __launch_bounds__(64) void ece_init(float* __restrict__ ws) {
    int i = threadIdx.x;
    if (i < 3 * NBINS) ws[i] = 0.0f;
}

// ---------------------------------------------------------------------------
// Kernel 2: one wave32 per row.
//   stage row -> LDS via GLOBAL_LOAD_ASYNC_TO_LDS_B128 (gfx1250 async path),
//   s_wait_asynccnt 0, then two LDS passes: (max, argmax) and sum(exp(x-m)),
//   wave32 shuffle reductions, LDS bin histogram, global atomic flush.
// ---------------------------------------------------------------------------
__global__ __launch_bounds__(BLOCK_THREADS) void ece_main(
    const float* __restrict__ input,
    const int*   __restrict__ target,
    float*       __restrict__ ws) {

    __shared__ float sbins[3 * NBINS];
    __shared__ float srow[WAVES_PER_BLOCK][ROW_PAD];

    const int tid = threadIdx.x;
    if (tid < 3 * NBINS) sbins[tid] = 0.0f;
    __syncthreads();

    const int lane = tid & 31;
    const int wave = tid >> 5;
    const int row  = blockIdx.x * WAVES_PER_BLOCK + wave;   // grid sized exactly

    const float* rp = input + (size_t)row * NCOLS;
    const float NEG_INF = -__builtin_inff();

    // ---- neutralize the 24-float pad (cols 1000..1023): lanes 26..31 ----
    if (lane >= 26) {
        *reinterpret_cast<v4f*>(&srow[wave][896 + lane * 4]) =
            (v4f){NEG_INF, NEG_INF, NEG_INF, NEG_INF};      // ds_store_b128
    }

    // ---- async DMA the row into this wave's LDS slice (read-once -> NT) ----
#pragma unroll
    for (int k = 0; k < 7; ++k) {
        uint32_t la = (uint32_t)(uintptr_t)&srow[wave][k * 128 + lane * 4];
        uint64_t ga = (uint64_t)(uintptr_t)(rp + k * 128 + lane * 4);
        asm volatile("global_load_async_to_lds_b128 %0, %1, off th:TH_LOAD_NT"
                     :: "v"(la), "v"(ga) : "memory");
    }
    if (lane < 26) {   // 104-float tail = 26 lanes x b128
        uint32_t la = (uint32_t)(uintptr_t)&srow[wave][896 + lane * 4];
        uint64_t ga = (uint64_t)(uintptr_t)(rp + 896 + lane * 4);
        asm volatile("global_load_async_to_lds_b128 %0, %1, off th:TH_LOAD_NT"
                     :: "v"(la), "v"(ga) : "memory");
    }
    asm volatile("s_wait_asynccnt 0x0" ::: "memory");  // wave-private slice: no barrier needed

    // ---- pass 1: lane-local max + argmax from LDS (cols ascend per lane) ----
    float m  = NEG_INF;
    int   am = 0;
#pragma unroll
    for (int k = 0; k < 8; ++k) {
        const int base = k * 128 + lane * 4;
        const v4f x = *reinterpret_cast<const v4f*>(&srow[wave][base]);  // ds_load_b128
        const float xs[4] = { x.x, x.y, x.z, x.w };
#pragma unroll
        for (int e = 0; e < 4; ++e) {
            if (xs[e] > m) { m = xs[e]; am = base + e; }
        }
    }

    // ---- wave32 reduction of (max, argmax), ties -> lowest column index ----
#pragma unroll
    for (int off = 16; off > 0; off >>= 1) {
        float om = __shfl_xor(m, off, 32);
        int   oa = __shfl_xor(am, off, 32);
        if (om > m || (om == m && oa < am)) { m = om; am = oa; }
    }

    // ---- pass 2: sum exp(x-m) via native v_exp_f32 (exp2, log2e prescale) ----
    const float L2E = 1.4426950408889634f;
    float s = 0.0f;
#pragma unroll
    for (int k = 0; k < 8; ++k) {
        const v4f x = *reinterpret_cast<const v4f*>(&srow[wave][k * 128 + lane * 4]);
        s += __builtin_amdgcn_exp2f((x.x - m) * L2E);   // pad cols: exp2(-inf)=0
        s += __builtin_amdgcn_exp2f((x.y - m) * L2E);
        s += __builtin_amdgcn_exp2f((x.z - m) * L2E);
        s += __builtin_amdgcn_exp2f((x.w - m) * L2E);
    }
#pragma unroll
    for (int off = 16; off > 0; off >>= 1) {
        s += __shfl_xor(s, off, 32);
    }

    // ---- bin this row (lane 0 of each wave) into LDS accumulators ----
    if (lane == 0) {
        float conf = 1.0f / s;                      // softmax max prob
        float acc  = (am == target[row]) ? 1.0f : 0.0f;
        int bin = (int)ceilf(conf * (float)NBINS) - 1;
        bin = bin < 0 ? 0 : (bin > NBINS - 1 ? NBINS - 1 : bin);
        atomicAdd(&sbins[bin],             1.0f);   // ds_add_f32
        atomicAdd(&sbins[NBINS + bin],     conf);
        atomicAdd(&sbins[2 * NBINS + bin], acc);
    }
    __syncthreads();

    // ---- flush non-empty per-block bins to global accumulators ----
    if (tid < 3 * NBINS) {
        float val = sbins[tid];
        if (val != 0.0f) atomicAdd(&ws[tid], val);  // global_atomic_add_f32
    }
}

// ---------------------------------------------------------------------------
// Kernel 3: 15-bin finalize -> scalar ECE loss
// ---------------------------------------------------------------------------
__global__ __launch_bounds__(32) void ece_fin(const float* __restrict__ ws,
                                              float* __restrict__ out) {
    if (threadIdx.x == 0) {
        float loss = 0.0f;
        for (int b = 0; b < NBINS; ++b) {
            float cnt = ws[b];
            float sc  = ws[NBINS + b];
            float sa  = ws[2 * NBINS + b];
            float safe = cnt > 1.0f ? cnt : 1.0f;
            float per  = fabsf(sc / safe - sa / safe) * (cnt / (float)NROWS);
            if (cnt > 0.0f) loss += per;
        }
        out[0] = loss;
    }
}

// ---------------------------------------------------------------------------
extern "C" void kernel_launch(void* const* d_in, const int* in_sizes, int n_in,
                              void* d_out, int out_size, void* d_ws, size_t ws_size,
                              hipStream_t stream) {
    (void)in_sizes; (void)n_in; (void)out_size; (void)ws_size;
    const float* input  = (const float*)d_in[0];
    const int*   target = (const int*)d_in[1];
    float*       ws     = (float*)d_ws;
    float*       out    = (float*)d_out;

    ece_init<<<1, 64, 0, stream>>>(ws);
    ece_main<<<NROWS / WAVES_PER_BLOCK, BLOCK_THREADS, 0, stream>>>(input, target, ws);
    ece_fin<<<1, 32, 0, stream>>>(ws, out);
}